// LMRA_8976481649277
// MI455X (gfx1250) — compile-verified
//
#include <hip/hip_runtime.h>
#include <math.h>

// ---------------------------------------------------------------------------
// Problem constants
// ---------------------------------------------------------------------------
#define BATCH 32
#define CHN   3
#define HW    262144            // 512*512  (= 1<<18)
#define NPIX  (BATCH * HW)      // 8388608 pixels (per channel)

// d_out layout: F_out[B,3,H,W] | delta[B,1,H,W] | gain[B,1,H,W] | gm[B,1,H,W]
#define OFF_DELTA (3 * NPIX)
#define OFF_GAIN  (4 * NPIX)
#define OFF_GM    (5 * NPIX)

// workspace layout (float units)
#define WS_P1   0                     // 1024 blocks * 2 floats (sum, sumsq of S_hat)
#define WS_ST1  2048                  // per-batch: u_x, inv_sigma  (2*32) ; [64] = thr
#define WS_THR  (WS_ST1 + 64)
#define WS_P2   2176                  // 2048 blocks * 12 floats
#define WS_ST2  (WS_P2 + 2048 * 12)   // scale, alpha[3], beta[3]

typedef __attribute__((ext_vector_type(2))) float v2f;
typedef __attribute__((ext_vector_type(8))) float v8f;

__device__ __forceinline__ float sigm(float z) { return 1.f / (1.f + __expf(-z)); }
__device__ __forceinline__ float tanh_e(float x) {
  float e = __expf(-2.f * x);
  return (1.f - e) / (1.f + e);
}
__device__ __forceinline__ float clamp01(float v) { return fminf(fmaxf(v, 0.f), 1.f); }

// ---------------------------------------------------------------------------
// Pass 1: per-batch sum / sumsq of S_hat.  1024 blocks (32 per batch), 256 thr.
// ---------------------------------------------------------------------------
__global__ void k_reduce_s(const float* __restrict__ S, float* __restrict__ ws) {
  int blk = blockIdx.x;
  int b = blk >> 5;            // batch
  int chunk = blk & 31;        // 8192-elem chunk within batch
  int base = b * HW + chunk * 8192;
  int t = threadIdx.x;
  float s0 = 0.f, s1 = 0.f;
#pragma unroll
  for (int k = 0; k < 32; ++k) {
    float v = S[base + t + k * 256];
    s0 += v;
    s1 += v * v;
  }
  __shared__ float r0[256], r1[256];
  r0[t] = s0; r1[t] = s1;
  __syncthreads();
  for (int off = 128; off > 0; off >>= 1) {
    if (t < off) { r0[t] += r0[t + off]; r1[t] += r1[t + off]; }
    __syncthreads();
  }
  if (t == 0) { ws[WS_P1 + 2 * blk] = r0[0]; ws[WS_P1 + 2 * blk + 1] = r1[0]; }
}

// ---------------------------------------------------------------------------
// Finalize 1: per-batch LN stats (on x = S*dw + db) + global threshold.
// One wave (32 threads), double precision.
// ---------------------------------------------------------------------------
__global__ void k_finalize1(float* __restrict__ ws,
                            const float* __restrict__ dc1w,
                            const float* __restrict__ dc1b) {
  __shared__ double bsum[32];
  int t = threadIdx.x;   // 32 threads; thread t owns batch t
  double s0 = 0.0, s1 = 0.0;
  for (int k = 0; k < 32; ++k) {
    s0 += (double)ws[WS_P1 + 2 * (t * 32 + k)];
    s1 += (double)ws[WS_P1 + 2 * (t * 32 + k) + 1];
  }
  const double N = (double)HW;
  double meanS = s0 / N;
  double varS  = s1 / N - meanS * meanS;
  double w = (double)dc1w[0], bb = (double)dc1b[0];
  double ux = w * meanS + bb;
  double vx = w * w * varS;
  ws[WS_ST1 + 2 * t]     = (float)ux;
  ws[WS_ST1 + 2 * t + 1] = (float)(1.0 / sqrt(vx + 1e-6));
  bsum[t] = s0;
  __syncthreads();
  if (t == 0) {
    double tot = 0.0;
    for (int k = 0; k < 32; ++k) tot += bsum[k];
    ws[WS_THR] = (float)(tot / ((double)BATCH * N) * 0.6);
  }
}

// ---------------------------------------------------------------------------
// Pass 2 (main): per-pixel delta / pre-scale gain / gm, plus block partials of
// {fg, sum(Y0*gm), sum(Y1*gm), S_c[3], M_cc'[6]}.  2048 blocks * 256 thr * 16.
// ---------------------------------------------------------------------------
__global__ void k_main(const float* __restrict__ I, const float* __restrict__ S,
                       const float* __restrict__ dc1w, const float* __restrict__ dc1b,
                       const float* __restrict__ lnw, const float* __restrict__ lnb,
                       float* __restrict__ out, float* __restrict__ ws) {
  int t = threadIdx.x;
  int base = blockIdx.x * 4096;            // 4096 pixels per block, same batch
  int b = base >> 18;
  float thr = ws[WS_THR];
  float ux  = ws[WS_ST1 + 2 * b];
  float inv = ws[WS_ST1 + 2 * b + 1];
  float dw = dc1w[0], db = dc1b[0];
  float lw = lnw[0],  lb = lnb[0];

  float acc[12];
#pragma unroll
  for (int j = 0; j < 12; ++j) acc[j] = 0.f;

#pragma unroll 4
  for (int k = 0; k < 16; ++k) {
    int g = base + t + k * 256;
    int s = g & (HW - 1);
    float sh = S[g];
    float x  = sh * dw + db;
    float z  = (x - ux) * inv * lw + lb;
    float delta = sigm(z);
    float gain  = 1.f + 0.6f * tanh_e((1.4f / 0.6f) * delta * sh);
    float gm    = (sh > thr) ? 1.f : 0.f;

    int ib = b * (3 * HW) + s;
    float i0 = I[ib], i1 = I[ib + HW], i2 = I[ib + 2 * HW];
    float y0l = 0.299f * i0 + 0.587f * i1 + 0.114f * i2;

    acc[0] += gm;
    acc[1] += y0l * gm;
    acc[2] += y0l * gain * gm;
    acc[3] += i0; acc[4] += i1; acc[5] += i2;
    acc[6] += i0 * i0; acc[7] += i0 * i1; acc[8] += i0 * i2;
    acc[9] += i1 * i1; acc[10] += i1 * i2; acc[11] += i2 * i2;

    out[OFF_DELTA + g] = delta;
    out[OFF_GAIN  + g] = gain;   // pre-scale; finalized in pass 3
    out[OFF_GM    + g] = gm;
  }

  __shared__ float sred[256];
#pragma unroll
  for (int j = 0; j < 12; ++j) {
    sred[t] = acc[j];
    __syncthreads();
    for (int off = 128; off > 0; off >>= 1) {
      if (t < off) sred[t] += sred[t + off];
      __syncthreads();
    }
    if (t == 0) ws[WS_P2 + blockIdx.x * 12 + j] = sred[0];
    __syncthreads();
  }
}

// ---------------------------------------------------------------------------
// Finalize 2: scale + folded BN affine (eps = alpha*y + beta).  256 thr, f64.
// ---------------------------------------------------------------------------
__global__ void k_finalize2(float* __restrict__ ws, const float* __restrict__ rc1w,
                            const float* __restrict__ bnw, const float* __restrict__ bnb) {
  int t = threadIdx.x;
  double acc[12];
#pragma unroll
  for (int j = 0; j < 12; ++j) acc[j] = 0.0;
  for (int r = t; r < 2048; r += 256)
#pragma unroll
    for (int j = 0; j < 12; ++j) acc[j] += (double)ws[WS_P2 + r * 12 + j];

  __shared__ double sd[256];
  __shared__ double res[12];
#pragma unroll
  for (int j = 0; j < 12; ++j) {
    sd[t] = acc[j];
    __syncthreads();
    for (int off = 128; off > 0; off >>= 1) {
      if (t < off) sd[t] += sd[t + off];
      __syncthreads();
    }
    if (t == 0) res[j] = sd[0];
    __syncthreads();
  }

  if (t == 0) {
    double fg = res[0];
    double mean0 = res[1] / (fg + 1e-6);
    double mean1 = res[2] / (fg + 1e-6);
    double target = mean0 * 1.25; if (target > 0.9) target = 0.9;
    double scale = target / (mean1 + 1e-6); if (scale > 1.0) scale = 1.0;
    if (!(fg > 0.0)) scale = 1.0;
    ws[WS_ST2] = (float)scale;

    const double N = (double)NPIX;
    double Sc[3] = {res[3], res[4], res[5]};
    double M[3][3];
    M[0][0] = res[6]; M[0][1] = M[1][0] = res[7]; M[0][2] = M[2][0] = res[8];
    M[1][1] = res[9]; M[1][2] = M[2][1] = res[10]; M[2][2] = res[11];
    for (int o = 0; o < 3; ++o) {
      double w0 = rc1w[o * 3], w1 = rc1w[o * 3 + 1], w2 = rc1w[o * 3 + 2];
      double mean = (w0 * Sc[0] + w1 * Sc[1] + w2 * Sc[2]) / N;
      double ey2 = (w0 * w0 * M[0][0] + w1 * w1 * M[1][1] + w2 * w2 * M[2][2] +
                    2.0 * (w0 * w1 * M[0][1] + w0 * w2 * M[0][2] + w1 * w2 * M[1][2])) / N;
      double var = ey2 - mean * mean;
      double invs = 1.0 / sqrt(var + 1e-5);
      ws[WS_ST2 + 1 + o] = (float)(0.02 * (double)bnw[o] * invs);            // alpha
      ws[WS_ST2 + 4 + o] = (float)(0.02 * ((double)bnb[o] - mean * (double)bnw[o] * invs)); // beta
    }
  }
}

// ---------------------------------------------------------------------------
// Pass 3 (final): gain_final, F_out with the 3x3 channel mix done on the
// matrix pipe via V_WMMA_F32_16X16X4_F32.  32768 blocks * 256 thr, 1 px/thr.
// A(16x4) = rc1_w zero-padded; B(4x16) = I channels for 16 pixels.
// f32 A layout: lanes0-15 -> {K0,K1}, lanes16-31 -> {K2,K3} (ISA 7.12.2).
// Grid covers NPIX exactly -> EXEC all-ones at the WMMAs.
// ---------------------------------------------------------------------------
__global__ void k_final(const float* __restrict__ I, const float* __restrict__ rc1w,
                        float* __restrict__ out, const float* __restrict__ ws) {
  int g = blockIdx.x * 256 + threadIdx.x;
  int b = g >> 18;
  int s = g & (HW - 1);
  int lane = threadIdx.x & 31;
  int m = lane & 15;

  // A operand: weights (row o = m, cols = input channel), zero-padded
  v2f a;
  float a0 = 0.f, a1 = 0.f;
  if (m < 3) {
    if (lane < 16) { a0 = rc1w[m * 3 + 0]; a1 = rc1w[m * 3 + 1]; }
    else           { a0 = rc1w[m * 3 + 2]; }
  }
  a[0] = a0; a[1] = a1;

  int ib = b * (3 * HW) + s;
  float i0 = I[ib], i1 = I[ib + HW], i2 = I[ib + 2 * HW];

  // B operands: b1 covers this wave's pixels 0..15, b2 covers pixels 16..31.
  int srcLo = (lane < 16) ? lane : (lane - 16);
  int srcHi = (lane < 16) ? (lane + 16) : lane;
  float i2_lo = __shfl(i2, srcLo);   // I2 of pixel (lane-16) for upper half
  float i0_hi = __shfl(i0, srcHi);   // I0 of pixel (lane+16) for lower half
  float i1_hi = __shfl(i1, srcHi);

  v2f b1, b2;
  b1[0] = (lane < 16) ? i0 : i2_lo;  b1[1] = (lane < 16) ? i1 : 0.f;
  b2[0] = (lane < 16) ? i0_hi : i2;  b2[1] = (lane < 16) ? i1_hi : 0.f;

  v8f c = {};
  v8f d1 = __builtin_amdgcn_wmma_f32_16x16x4_f32(false, a, false, b1, (short)0, c, false, false);
  v8f d2 = __builtin_amdgcn_wmma_f32_16x16x4_f32(false, a, false, b2, (short)0, c, false, false);

  // lane p<16 reads y[o,p] from d1[o]; lane p>=16 pulls its pixel from d2 of lane p-16
  float t0 = __shfl(d2[0], m), t1 = __shfl(d2[1], m), t2 = __shfl(d2[2], m);
  float y0 = (lane < 16) ? d1[0] : t0;
  float y1 = (lane < 16) ? d1[1] : t1;
  float y2 = (lane < 16) ? d1[2] : t2;

  float scale = ws[WS_ST2];
  float al0 = ws[WS_ST2 + 1], al1 = ws[WS_ST2 + 2], al2 = ws[WS_ST2 + 3];
  float be0 = ws[WS_ST2 + 4], be1 = ws[WS_ST2 + 5], be2 = ws[WS_ST2 + 6];

  float gp = out[OFF_GAIN + g];
  float gf = 1.f + (gp - 1.f) * scale;
  out[OFF_GAIN + g] = gf;

  out[ib]          = clamp01(i0 * gf + (al0 * y0 + be0));
  out[ib + HW]     = clamp01(i1 * gf + (al1 * y1 + be1));
  out[ib + 2 * HW] = clamp01(i2 * gf + (al2 * y2 + be2));
}

// ---------------------------------------------------------------------------
extern "C" void kernel_launch(void* const* d_in, const int* in_sizes, int n_in,
                              void* d_out, int out_size, void* d_ws, size_t ws_size,
                              hipStream_t stream) {
  const float* I    = (const float*)d_in[0];
  const float* S    = (const float*)d_in[1];
  const float* dc1w = (const float*)d_in[2];
  const float* dc1b = (const float*)d_in[3];
  const float* lnw  = (const float*)d_in[4];
  const float* lnb  = (const float*)d_in[5];
  const float* rc1w = (const float*)d_in[6];
  const float* bnw  = (const float*)d_in[7];
  const float* bnb  = (const float*)d_in[8];
  float* out = (float*)d_out;
  float* ws  = (float*)d_ws;

  k_reduce_s <<<1024,  256, 0, stream>>>(S, ws);
  k_finalize1<<<1,      32, 0, stream>>>(ws, dc1w, dc1b);
  k_main     <<<2048,  256, 0, stream>>>(I, S, dc1w, dc1b, lnw, lnb, out, ws);
  k_finalize2<<<1,     256, 0, stream>>>(ws, rc1w, bnw, bnb);
  k_final    <<<32768, 256, 0, stream>>>(I, rc1w, out, ws);
}